// SlowStrategicReasoner_10806137717192
// MI455X (gfx1250) — compile-verified
//
#include <hip/hip_runtime.h>
#include <hip/hip_bf16.h>

typedef __attribute__((ext_vector_type(2))) float v2f;
typedef __attribute__((ext_vector_type(8))) float v8f;

#define NB    32      // batch
#define SEQ   2048
#define DIM   512
#define HID   256
#define ND    128
#define NN    128     // MAX_NODES

// padded LDS strides (stride mod 32 == 4 -> conflict-free 16-lane column reads,
// stride*4 bytes % 16 == 0 -> float4-aligned rows)
#define APAD  516     // 512+4
#define HPAD  260     // 256+4
#define TPAD  132     // 128+4
#define UPAD  68      // 64+4
#define SPAD  36      // 32+4

static __device__ __forceinline__ v8f wmma_f32(v2f a, v2f b, v8f c) {
  return __builtin_amdgcn_wmma_f32_16x16x4_f32(false, a, false, b, (short)0, c, false, false);
}

// ---------------------------------------------------------------------------
// Kernel 1: gather + encoder MLP (512->256 relu, 256->128) + layernorm -> nf
// grid: 256 blocks (32 batches x 8 node-tiles), 256 threads (8 waves)
// ---------------------------------------------------------------------------
__global__ void __launch_bounds__(256)
ssr_encoder(const float* __restrict__ x, const float* __restrict__ w1,
            const float* __restrict__ b1, const float* __restrict__ w2,
            const float* __restrict__ b2, const float* __restrict__ g,
            const float* __restrict__ be, float* __restrict__ nf) {
  __shared__ float Alds[16 * APAD];
  __shared__ float Hlds[16 * HPAD];
  __shared__ float Tlds[16 * TPAD];

  const int b = blockIdx.x >> 3;
  const int itile = blockIdx.x & 7;
  const int tid = threadIdx.x;

  // gather 16 node rows (512 f32 each) into LDS with float4 loads
  for (int q = tid; q < 16 * 128; q += 256) {
    const int r  = q >> 7;
    const int c4 = (q & 127) << 2;
    const int node = itile * 16 + r;
    const int srow = (int)((double)node * (2047.0 / 127.0));   // linspace idx
    const float4 v = *(const float4*)(x + ((size_t)b * SEQ + srow) * DIM + c4);
    *(float4*)(&Alds[r * APAD + c4]) = v;
  }
  __syncthreads();

  const int wave = tid >> 5;
  const int lane = tid & 31;
  const int n    = lane & 15;
  const int half = lane >> 4;
  const int kh   = half << 1;            // f32 16x16x4 A/B fragment K pair
  const int mbase = half << 3;

  // stage 1: h = relu(A @ w1 + b1), each wave owns 2 of 16 column tiles
#pragma unroll 1
  for (int t = 0; t < 2; ++t) {
    const int cb = wave * 32 + t * 16;
    v8f acc = {};
#pragma unroll 4
    for (int kk = 0; kk < DIM; kk += 4) {
      v2f a, bf;
      a.x = Alds[n * APAD + kk + kh];
      a.y = Alds[n * APAD + kk + kh + 1];
      bf.x = w1[(size_t)(kk + kh) * HID + cb + n];
      bf.y = w1[(size_t)(kk + kh + 1) * HID + cb + n];
      acc = wmma_f32(a, bf, acc);
    }
#pragma unroll
    for (int r = 0; r < 8; ++r) {
      const int m = mbase + r;
      const int col = cb + n;
      Hlds[m * HPAD + col] = fmaxf(acc[r] + b1[col], 0.0f);
    }
  }
  __syncthreads();

  // stage 2: t = h @ w2 + b2, each wave owns one of 8 column tiles
  {
    const int cb = wave * 16;
    v8f acc = {};
#pragma unroll 4
    for (int kk = 0; kk < HID; kk += 4) {
      v2f a, bf;
      a.x = Hlds[n * HPAD + kk + kh];
      a.y = Hlds[n * HPAD + kk + kh + 1];
      bf.x = w2[(size_t)(kk + kh) * ND + cb + n];
      bf.y = w2[(size_t)(kk + kh + 1) * ND + cb + n];
      acc = wmma_f32(a, bf, acc);
    }
#pragma unroll
    for (int r = 0; r < 8; ++r) {
      const int m = mbase + r;
      const int col = cb + n;
      Tlds[m * TPAD + col] = acc[r] + b2[col];
    }
  }
  __syncthreads();

  // stage 3: layernorm over 128 features, one thread per row
  if (tid < 16) {
    float mean = 0.0f;
    for (int c = 0; c < ND; ++c) mean += Tlds[tid * TPAD + c];
    mean *= (1.0f / ND);
    float var = 0.0f;
    for (int c = 0; c < ND; ++c) {
      const float d = Tlds[tid * TPAD + c] - mean;
      var += d * d;
    }
    var *= (1.0f / ND);
    const float rstd = rsqrtf(var + 1e-5f);
    const int node = itile * 16 + tid;
    float* o = nf + ((size_t)b * NN + node) * ND;
    for (int c = 0; c < ND; ++c)
      o[c] = (Tlds[tid * TPAD + c] - mean) * rstd * g[c] + be[c];
  }
}

// ---------------------------------------------------------------------------
// Kernel 2: projections U,V (nf @ ew1 halves) and SU,SV (nf @ sw1 halves)
// grid: 256 blocks, 128 threads (4 waves); 12 column tiles of 16 round-robin
// ---------------------------------------------------------------------------
__global__ void __launch_bounds__(128)
ssr_proj(const float* __restrict__ nf, const float* __restrict__ ew1,
         const float* __restrict__ sw1, float* __restrict__ U,
         float* __restrict__ V, float* __restrict__ SU, float* __restrict__ SV) {
  __shared__ float Alds[16 * TPAD];

  const int b = blockIdx.x >> 3;
  const int itile = blockIdx.x & 7;
  const int tid = threadIdx.x;

  for (int q = tid; q < 16 * 32; q += 128) {
    const int r  = q >> 5;
    const int c4 = (q & 31) << 2;
    *(float4*)(&Alds[r * TPAD + c4]) =
        *(const float4*)(nf + ((size_t)b * NN + itile * 16 + r) * ND + c4);
  }
  __syncthreads();

  const int wave = tid >> 5;
  const int lane = tid & 31;
  const int n    = lane & 15;
  const int half = lane >> 4;
  const int kh   = half << 1;
  const int mbase = half << 3;

  for (int tt = wave; tt < 12; tt += 4) {
    const float* W;
    float* O;
    int cb, ncols;
    if (tt < 4)       { W = ew1;              O = U;  cb = tt * 16;        ncols = 64; }
    else if (tt < 8)  { W = ew1 + 128 * 64;   O = V;  cb = (tt - 4) * 16;  ncols = 64; }
    else if (tt < 10) { W = sw1;              O = SU; cb = (tt - 8) * 16;  ncols = 32; }
    else              { W = sw1 + 128 * 32;   O = SV; cb = (tt - 10) * 16; ncols = 32; }

    v8f acc = {};
#pragma unroll 4
    for (int kk = 0; kk < ND; kk += 4) {
      v2f a, bf;
      a.x = Alds[n * TPAD + kk + kh];
      a.y = Alds[n * TPAD + kk + kh + 1];
      bf.x = W[(size_t)(kk + kh) * ncols + cb + n];
      bf.y = W[(size_t)(kk + kh + 1) * ncols + cb + n];
      acc = wmma_f32(a, bf, acc);
    }
#pragma unroll
    for (int r = 0; r < 8; ++r) {
      const int m = mbase + r;
      O[((size_t)b * NN + itile * 16 + m) * ncols + cb + n] = acc[r];
    }
  }
}

// ---------------------------------------------------------------------------
// Kernel 3: pairwise edge + strength heads
// grid: 256 blocks (batch x i-tile), 256 threads; wave w owns j-tile w
// ---------------------------------------------------------------------------
__global__ void __launch_bounds__(256)
ssr_pairs(const float* __restrict__ U, const float* __restrict__ V,
          const float* __restrict__ SU, const float* __restrict__ SV,
          const float* __restrict__ eb1, const float* __restrict__ ew2,
          const float* __restrict__ eb2, const float* __restrict__ ew3,
          const float* __restrict__ eb3, const float* __restrict__ sb1,
          const float* __restrict__ sw2, const float* __restrict__ sb2,
          float* __restrict__ adj, float* __restrict__ str) {
  __shared__ float Ul[16 * UPAD];
  __shared__ float Vl[128 * UPAD];
  __shared__ float SUl[16 * SPAD];
  __shared__ float SVl[128 * SPAD];
  __shared__ float eb1l[64], eb2l[32], w3l[32], sb1l[32], sw2l[32];

  const int b = blockIdx.x >> 3;
  const int itile = blockIdx.x & 7;
  const int i0 = itile * 16;
  const int tid = threadIdx.x;

  // cooperative staged loads (float4, padded-stride LDS)
  {
    const int q = tid;                                    // Ul: 256 f4
    const int r = q >> 4, c4 = (q & 15) << 2;
    *(float4*)(&Ul[r * UPAD + c4]) =
        *(const float4*)(U + ((size_t)b * NN + i0 + r) * 64 + c4);
  }
  for (int q = tid; q < 2048; q += 256) {                 // Vl: 2048 f4
    const int r = q >> 4, c4 = (q & 15) << 2;
    *(float4*)(&Vl[r * UPAD + c4]) =
        *(const float4*)(V + ((size_t)b * NN + r) * 64 + c4);
  }
  if (tid < 128) {                                        // SUl: 128 f4
    const int r = tid >> 3, c4 = (tid & 7) << 2;
    *(float4*)(&SUl[r * SPAD + c4]) =
        *(const float4*)(SU + ((size_t)b * NN + i0 + r) * 32 + c4);
  }
  for (int q = tid; q < 1024; q += 256) {                 // SVl: 1024 f4
    const int r = q >> 3, c4 = (q & 7) << 2;
    *(float4*)(&SVl[r * SPAD + c4]) =
        *(const float4*)(SV + ((size_t)b * NN + r) * 32 + c4);
  }
  if (tid < 64) eb1l[tid] = eb1[tid];
  if (tid < 32) {
    eb2l[tid] = eb2[tid];
    w3l[tid]  = ew3[tid];
    sb1l[tid] = sb1[tid];
    sw2l[tid] = sw2[tid];
  }
  __syncthreads();

  const int wave = tid >> 5;
  const int lane = tid & 31;
  const int n    = lane & 15;
  const int half = lane >> 4;
  const int kh   = half << 1;
  const int jt   = wave;                // 8 waves cover all 128 j
  const float eb3v = eb3[0];
  const float sb2v = sb2[0];

  // hoist ew2 B-fragments (K=64 -> 16 steps, two 16-col tiles)
  v2f bE0[16], bE1[16];
#pragma unroll
  for (int t = 0; t < 16; ++t) {
    const int k0 = 4 * t + kh;
    bE0[t].x = ew2[k0 * 32 + n];
    bE0[t].y = ew2[(k0 + 1) * 32 + n];
    bE1[t].x = ew2[k0 * 32 + 16 + n];
    bE1[t].y = ew2[(k0 + 1) * 32 + 16 + n];
  }

#pragma unroll 1
  for (int i = 0; i < 16; ++i) {
    const int gi = i0 + i;
    v8f acc0 = {}, acc1 = {};
#pragma unroll
    for (int t = 0; t < 16; ++t) {
      const int k0 = 4 * t + kh;
      const float u0 = Ul[i * UPAD + k0];
      const float u1 = Ul[i * UPAD + k0 + 1];
      const float vv0 = Vl[(jt * 16 + n) * UPAD + k0];
      const float vv1 = Vl[(jt * 16 + n) * UPAD + k0 + 1];
      v2f a;
      a.x = fmaxf(u0 + vv0 + eb1l[k0], 0.0f);
      a.y = fmaxf(u1 + vv1 + eb1l[k0 + 1], 0.0f);
      acc0 = wmma_f32(a, bE0[t], acc0);
      acc1 = wmma_f32(a, bE1[t], acc1);
    }
    // adj head: relu(+eb2) dot ew3, reduce across 16 lanes of each half
#pragma unroll
    for (int r = 0; r < 8; ++r) {
      float pa = fmaxf(acc0[r] + eb2l[n], 0.0f) * w3l[n] +
                 fmaxf(acc1[r] + eb2l[16 + n], 0.0f) * w3l[16 + n];
      pa += __shfl_xor(pa, 1);
      pa += __shfl_xor(pa, 2);
      pa += __shfl_xor(pa, 4);
      pa += __shfl_xor(pa, 8);
      const int m = r + (half << 3);
      const int j = jt * 16 + m;
      if (n == 0)
        adj[((size_t)b * NN + gi) * NN + j] =
            (gi == j) ? 0.0f : 1.0f / (1.0f + expf(-(pa + eb3v)));
    }
    // strength head (32-wide): half-wave split dot, combine across halves
    float ss = 0.0f;
#pragma unroll
    for (int k = 0; k < 16; ++k) {
      const int kk = half * 16 + k;
      ss += fmaxf(SUl[i * SPAD + kk] + SVl[(jt * 16 + n) * SPAD + kk] + sb1l[kk],
                  0.0f) * sw2l[kk];
    }
    ss += __shfl_xor(ss, 16);
    const int j = jt * 16 + n;
    if (half == 0)
      str[((size_t)b * NN + gi) * NN + j] =
          (gi == j) ? 0.0f : tanhf(ss + sb2v);
  }
}

// ---------------------------------------------------------------------------
extern "C" void kernel_launch(void* const* d_in, const int* in_sizes, int n_in,
                              void* d_out, int out_size, void* d_ws, size_t ws_size,
                              hipStream_t stream) {
  const float* x      = (const float*)d_in[0];
  const float* enc_w1 = (const float*)d_in[1];
  const float* enc_b1 = (const float*)d_in[2];
  const float* enc_w2 = (const float*)d_in[3];
  const float* enc_b2 = (const float*)d_in[4];
  const float* ln_g   = (const float*)d_in[5];
  const float* ln_b   = (const float*)d_in[6];
  const float* ew1    = (const float*)d_in[7];
  const float* eb1    = (const float*)d_in[8];
  const float* ew2    = (const float*)d_in[9];
  const float* eb2    = (const float*)d_in[10];
  const float* ew3    = (const float*)d_in[11];
  const float* eb3    = (const float*)d_in[12];
  const float* sw1    = (const float*)d_in[13];
  const float* sb1    = (const float*)d_in[14];
  const float* sw2    = (const float*)d_in[15];
  const float* sb2    = (const float*)d_in[16];

  float* out = (float*)d_out;
  float* nf  = out;                       // 32*128*128
  float* adj = out + (size_t)NB * NN * ND;
  float* str = adj + (size_t)NB * NN * NN;

  float* ws = (float*)d_ws;
  float* U  = ws;                         // 32*128*64
  float* V  = U + (size_t)NB * NN * 64;
  float* SU = V + (size_t)NB * NN * 64;   // 32*128*32
  float* SV = SU + (size_t)NB * NN * 32;

  ssr_encoder<<<dim3(NB * 8), dim3(256), 0, stream>>>(
      x, enc_w1, enc_b1, enc_w2, enc_b2, ln_g, ln_b, nf);
  ssr_proj<<<dim3(NB * 8), dim3(128), 0, stream>>>(
      nf, ew1, sw1, U, V, SU, SV);
  ssr_pairs<<<dim3(NB * 8), dim3(256), 0, stream>>>(
      U, V, SU, SV, eb1, ew2, eb2, ew3, eb3, sb1, sw2, sb2, adj, str);
}